// PillarFeatureNet_52561809768795
// MI455X (gfx1250) — compile-verified
//
#include <hip/hip_runtime.h>
#include <hip/hip_bf16.h>
#include <math.h>

typedef __attribute__((ext_vector_type(2))) float v2f;
typedef __attribute__((ext_vector_type(8))) float v8f;

#define T_PTS 32
#define C_IN  4
#define AUG   8   // 7 real features, padded to 8 (K padding for WMMA)
#define H1C   32
#define H2C   64
#define WAVES 8
#define BLOCK (WAVES * 32)

// ws layout (floats)
#define WS_GS1 0
#define WS_GQ1 32
#define WS_SC1 64
#define WS_SH1 96
#define WS_GS2 128
#define WS_GQ2 192
#define WS_SC2 256
#define WS_SH2 320
#define WS_TOT 384

__device__ __forceinline__ float wave_sum32(float v) {
#pragma unroll
    for (int m = 16; m >= 1; m >>= 1) v += __shfl_xor(v, m, 32);
    return v;
}

__device__ __forceinline__ float silu_f(float x) {
    return x / (1.0f + __expf(-x));
}

// MODE 1: layer1 stats; MODE 2: layer2 stats; MODE 3: final output
template <int MODE>
__global__ void __launch_bounds__(BLOCK)
pfn_kernel(const float* __restrict__ pillars,
           const float* __restrict__ W1,
           const float* __restrict__ W2,
           const int*   __restrict__ num_points,
           float*       __restrict__ ws,
           float*       __restrict__ out,
           int P)
{
    __shared__ float sW1[AUG][H1C];              // 1 KB  (row 7 zero-padded)
    __shared__ float sW2[H1C][H2C];              // 8 KB
    __shared__ float sX [WAVES][T_PTS][AUG];     // 8 KB  (per-wave augmented features)
    __shared__ float sH1[WAVES][T_PTS][H1C];     // 32 KB (per-wave layer1 activations)
    __shared__ float sStat[2 * H2C];             // block-level sum / sumsq

    const int tid  = threadIdx.x;
    const int w    = tid >> 5;
    const int lane = tid & 31;
    const int half = lane >> 4;   // 0: lanes 0-15, 1: lanes 16-31
    const int lm   = lane & 15;

    // ---- stage weights + zero block stats ----
    {
        int k = tid >> 5, n = tid & 31;          // 256 threads = 8x32 exactly
        sW1[k][n] = (k < 7) ? W1[k * H1C + n] : 0.0f;
    }
    for (int i = tid; i < H1C * H2C; i += BLOCK) sW2[i >> 6][i & 63] = W2[i];
    if (tid < 2 * H2C) sStat[tid] = 0.0f;

    // ---- per-wave pillar features (lane == point index t) ----
    const int  p      = blockIdx.x * WAVES + w;
    const bool active = (p < P);
    const int  pc     = active ? p : 0;

    const float4 v = ((const float4*)pillars)[(size_t)pc * T_PTS + lane];
    const int np   = num_points[pc];
    const float m  = (active && lane < np) ? 1.0f : 0.0f;
    const float cnt = (np > 0) ? (float)np : 1.0f;
    const float cx = wave_sum32(m * v.x) / cnt;
    const float cy = wave_sum32(m * v.y) / cnt;

    sX[w][lane][0] = m * v.x;
    sX[w][lane][1] = m * v.y;
    sX[w][lane][2] = m * v.z;
    sX[w][lane][3] = m * v.w;
    sX[w][lane][4] = m * (v.x - cx);
    sX[w][lane][5] = m * (v.y - cy);
    sX[w][lane][6] = m * sqrtf(v.x * v.x + v.y * v.y + v.z * v.z);
    sX[w][lane][7] = 0.0f;

    __syncthreads();

    // ---- layer 1: (32 x 7) @ (7 x 32) via 16x16x4 fp32 WMMA ----
    // A frags: lanes 0-15 -> M=lm, K={k0,k0+1}; lanes 16-31 -> same M, K+2
    v2f a1[2][2];
#pragma unroll
    for (int mt = 0; mt < 2; ++mt)
#pragma unroll
        for (int kc = 0; kc < 2; ++kc) {
            int row = mt * 16 + lm;
            int k0  = kc * 4 + half * 2;
            a1[mt][kc] = v2f{sX[w][row][k0], sX[w][row][k0 + 1]};
        }
    v2f b1f[2][2];
#pragma unroll
    for (int kc = 0; kc < 2; ++kc)
#pragma unroll
        for (int nt = 0; nt < 2; ++nt) {
            int k0 = kc * 4 + half * 2;
            int n  = nt * 16 + lm;
            b1f[kc][nt] = v2f{sW1[k0][n], sW1[k0 + 1][n]};
        }

    v8f h1acc[2][2];
#pragma unroll
    for (int mt = 0; mt < 2; ++mt)
#pragma unroll
        for (int nt = 0; nt < 2; ++nt) {
            v8f acc = {0.f, 0.f, 0.f, 0.f, 0.f, 0.f, 0.f, 0.f};
#pragma unroll
            for (int kc = 0; kc < 2; ++kc)
                acc = __builtin_amdgcn_wmma_f32_16x16x4_f32(
                    false, a1[mt][kc], false, b1f[kc][nt], (short)0, acc, false, false);
            h1acc[mt][nt] = acc;
        }

    if constexpr (MODE == 1) {
        // per-channel sum / sumsq of pre-BN layer1 over this pillar's 32 rows
        float s[2] = {0.f, 0.f}, q[2] = {0.f, 0.f};
#pragma unroll
        for (int mt = 0; mt < 2; ++mt)
#pragma unroll
            for (int nt = 0; nt < 2; ++nt)
#pragma unroll
                for (int r = 0; r < 8; ++r) {
                    float x = h1acc[mt][nt][r];
                    s[nt] += x; q[nt] += x * x;
                }
#pragma unroll
        for (int nt = 0; nt < 2; ++nt) {
            s[nt] += __shfl_xor(s[nt], 16, 32);
            q[nt] += __shfl_xor(q[nt], 16, 32);
            if (active && half == 0) {
                atomicAdd(&sStat[nt * 16 + lm], s[nt]);
                atomicAdd(&sStat[H2C + nt * 16 + lm], q[nt]);
            }
        }
        __syncthreads();
        if (tid < H1C) {
            atomicAdd(&ws[WS_GS1 + tid], sStat[tid]);
            atomicAdd(&ws[WS_GQ1 + tid], sStat[H2C + tid]);
        }
        return;
    }

    // ---- BN1 + SiLU, then layer 2 ----
    float sc1[2], sh1[2];
#pragma unroll
    for (int nt = 0; nt < 2; ++nt) {
        sc1[nt] = ws[WS_SC1 + nt * 16 + lm];
        sh1[nt] = ws[WS_SH1 + nt * 16 + lm];
    }
    float sc2[4], sh2[4];
    if constexpr (MODE == 3) {
#pragma unroll
        for (int nt = 0; nt < 4; ++nt) {
            sc2[nt] = ws[WS_SC2 + nt * 16 + lm];
            sh2[nt] = ws[WS_SH2 + nt * 16 + lm];
        }
    }

    float s[4] = {0.f, 0.f, 0.f, 0.f}, q[4] = {0.f, 0.f, 0.f, 0.f};
    float mx[4] = {-3.0e38f, -3.0e38f, -3.0e38f, -3.0e38f};

#pragma unroll
    for (int mt = 0; mt < 2; ++mt) {
        // stage h1 = silu(bn(h1_pre)) rows [mt*16, mt*16+16) into LDS
        // D layout: VGPR r -> row = mt*16 + half*8 + r, col = nt*16 + lm
#pragma unroll
        for (int nt = 0; nt < 2; ++nt)
#pragma unroll
            for (int r = 0; r < 8; ++r) {
                float h = silu_f(h1acc[mt][nt][r] * sc1[nt] + sh1[nt]);
                sH1[w][mt * 16 + half * 8 + r][nt * 16 + lm] = h;
            }
        // layer2: (16 x 32) @ (32 x 64), K chunks of 4
        v8f acc2[4];
#pragma unroll
        for (int nt = 0; nt < 4; ++nt) acc2[nt] = v8f{0.f, 0.f, 0.f, 0.f, 0.f, 0.f, 0.f, 0.f};
#pragma unroll
        for (int kc = 0; kc < 8; ++kc) {
            int k0 = kc * 4 + half * 2;
            v2f a2 = v2f{sH1[w][mt * 16 + lm][k0], sH1[w][mt * 16 + lm][k0 + 1]};
#pragma unroll
            for (int nt = 0; nt < 4; ++nt) {
                v2f b2v = v2f{sW2[k0][nt * 16 + lm], sW2[k0 + 1][nt * 16 + lm]};
                acc2[nt] = __builtin_amdgcn_wmma_f32_16x16x4_f32(
                    false, a2, false, b2v, (short)0, acc2[nt], false, false);
            }
        }
        if constexpr (MODE == 2) {
#pragma unroll
            for (int nt = 0; nt < 4; ++nt)
#pragma unroll
                for (int r = 0; r < 8; ++r) {
                    float x = acc2[nt][r];
                    s[nt] += x; q[nt] += x * x;
                }
        } else {
#pragma unroll
            for (int nt = 0; nt < 4; ++nt)
#pragma unroll
                for (int r = 0; r < 8; ++r) {
                    float h = silu_f(acc2[nt][r] * sc2[nt] + sh2[nt]);
                    int row = mt * 16 + half * 8 + r;   // point index t
                    float val = (row < np) ? h : 0.0f;  // mask -> 0 (matches feats*mask)
                    mx[nt] = fmaxf(mx[nt], val);
                }
        }
    }

    if constexpr (MODE == 2) {
#pragma unroll
        for (int nt = 0; nt < 4; ++nt) {
            s[nt] += __shfl_xor(s[nt], 16, 32);
            q[nt] += __shfl_xor(q[nt], 16, 32);
            if (active && half == 0) {
                atomicAdd(&sStat[nt * 16 + lm], s[nt]);
                atomicAdd(&sStat[H2C + nt * 16 + lm], q[nt]);
            }
        }
        __syncthreads();
        if (tid < H2C) {
            atomicAdd(&ws[WS_GS2 + tid], sStat[tid]);
            atomicAdd(&ws[WS_GQ2 + tid], sStat[H2C + tid]);
        }
    } else {
#pragma unroll
        for (int nt = 0; nt < 4; ++nt) {
            mx[nt] = fmaxf(mx[nt], __shfl_xor(mx[nt], 16, 32));
            if (active && half == 0)
                out[(size_t)p * H2C + nt * 16 + lm] = mx[nt];
        }
    }
}

__global__ void pfn_zero(float* __restrict__ ws) {
    if (threadIdx.x < WS_TOT) ws[threadIdx.x] = 0.0f;
}

__global__ void pfn_finalize(const float* __restrict__ g, const float* __restrict__ b,
                             float* __restrict__ ws, int gs, int gq, int sc, int sh,
                             int C, float invN) {
    int c = threadIdx.x;
    if (c < C) {
        float mu   = ws[gs + c] * invN;
        float var  = ws[gq + c] * invN - mu * mu;   // biased variance (matches reference)
        float rstd = rsqrtf(var + 1e-3f);
        float scv  = g[c] * rstd;
        ws[sc + c] = scv;
        ws[sh + c] = b[c] - mu * scv;
    }
}

extern "C" void kernel_launch(void* const* d_in, const int* in_sizes, int n_in,
                              void* d_out, int out_size, void* d_ws, size_t ws_size,
                              hipStream_t stream) {
    const float* pillars = (const float*)d_in[0];
    const float* W1      = (const float*)d_in[1];
    const float* g1      = (const float*)d_in[2];
    const float* b1      = (const float*)d_in[3];
    const float* W2      = (const float*)d_in[4];
    const float* g2      = (const float*)d_in[5];
    const float* b2      = (const float*)d_in[6];
    const int*   np      = (const int*)d_in[7];
    float* out = (float*)d_out;
    float* ws  = (float*)d_ws;

    const int P = in_sizes[7];                       // num_points has P elements
    const int blocks = (P + WAVES - 1) / WAVES;      // one pillar per wave
    const float invN = 1.0f / ((float)P * (float)T_PTS);

    pfn_zero<<<1, WS_TOT, 0, stream>>>(ws);
    pfn_kernel<1><<<blocks, BLOCK, 0, stream>>>(pillars, W1, W2, np, ws, out, P);
    pfn_finalize<<<1, 64, 0, stream>>>(g1, b1, ws, WS_GS1, WS_GQ1, WS_SC1, WS_SH1, H1C, invN);
    pfn_kernel<2><<<blocks, BLOCK, 0, stream>>>(pillars, W1, W2, np, ws, out, P);
    pfn_finalize<<<1, 64, 0, stream>>>(g2, b2, ws, WS_GS2, WS_GQ2, WS_SC2, WS_SH2, H2C, invN);
    pfn_kernel<3><<<blocks, BLOCK, 0, stream>>>(pillars, W1, W2, np, ws, out, P);
}